// CrossEntropy2dWeighted_20598663151778
// MI455X (gfx1250) — compile-verified
//
#include <hip/hip_runtime.h>
#include <stdint.h>

// ---------------------------------------------------------------------------
// Class-balanced BCE ("CrossEntropy2dWeighted") for MI455X / gfx1250.
// Bandwidth-bound (335 MB in / 1 float out, ~14us at 23.3 TB/s). Uses the
// CDNA5 Tensor Data Mover to stream predict tiles into LDS (double-buffered,
// TENSORcnt-synchronized) while wave32 waves compute -log terms out of LDS.
// ---------------------------------------------------------------------------

#define CE_N 8
#define CE_C 19
#define CE_H 512
#define CE_W 1024
#define CE_HW (CE_H * CE_W)                 // 524288 = 2^19
#define CE_NPIX (CE_N * CE_HW)              // 4194304
#define CE_TOTAL (CE_N * CE_C * CE_HW)      // 79691776

#define TILE 4096                           // floats per TDM tile (16 KB)
#define K_TILES 16                          // tiles per block
#define THREADS 256                         // 8 wave32 waves
#define MAIN_BLOCKS (CE_TOTAL / (TILE * K_TILES))  // 1216

#define HAVE_TDM 0
#if defined(__has_builtin)
#if __has_builtin(__builtin_amdgcn_tensor_load_to_lds) && \
    __has_builtin(__builtin_amdgcn_s_wait_tensorcnt)
#undef HAVE_TDM
#define HAVE_TDM 1
#endif
#endif

typedef __attribute__((ext_vector_type(4))) unsigned int tdm_u32x4;
typedef __attribute__((ext_vector_type(8))) int          tdm_i32x8;
typedef __attribute__((ext_vector_type(4))) int          tdm_i32x4;

struct WS {                       // workspace layout in d_ws
  double             sum;         // weighted BCE accumulator
  unsigned long long pos;         // #(t > 0)
  unsigned long long neg;         // #(t == 0)
  unsigned long long maskc;       // #(t >= 0 && t != 255)
};

// ---------------------------------------------------------------------------
// TDM descriptor per CDNA5 ISA ch.8: 1-D tile of `nelem` 4-byte elements.
// group0: count=1 | lds_addr | global_addr[56:0] | type=2
// group1: data_size=4B, tensor_dim0=nelem, tensor_dim1=1, tile_dim0=nelem,
//         tensor_dim0_stride=nelem; no multicast/pad/iterate/barrier.
// Toolchain here exposes the 6-arg builtin:
//   (uint32x4 g0, int32x8 g1, int32x4 g2, int32x4 g3, int32x8 g4, i32 cpol)
// ---------------------------------------------------------------------------
#if HAVE_TDM
__device__ __forceinline__ void tdm_issue_1d(unsigned lds_byte,
                                             const float* gptr, int nelem) {
  unsigned long long ga = (unsigned long long)(uintptr_t)gptr;
  tdm_u32x4 g0;
  g0.x = 1u;                                             // count = 1 (valid D#)
  g0.y = lds_byte;                                       // LDS byte address
  g0.z = (unsigned)(ga & 0xFFFFFFFFu);                   // global_addr[31:0]
  g0.w = (unsigned)((ga >> 32) & 0x01FFFFFFu)            // global_addr[56:32]
       | (2u << 30);                                     // type = 2 ("image")
  tdm_i32x8 g1;
  g1[0] = (int)(2u << 16);                               // data_size = 2 (4 B)
  g1[1] = (int)(((unsigned)nelem & 0xFFFFu) << 16);      // tensor_dim0[15:0]
  g1[2] = (int)(((unsigned)nelem >> 16) | (1u << 16));   // dim0[31:16], dim1=1
  g1[3] = (int)(((unsigned)nelem & 0xFFFFu) << 16);      // tile_dim0
  g1[4] = 0;                                             // tile_dim1/2 unused
  g1[5] = nelem;                                         // dim0_stride[31:0]
  g1[6] = 0;
  g1[7] = 0;
  tdm_i32x4 gz4 = {0, 0, 0, 0};                          // <=2D: groups 2/3 off
  tdm_i32x8 gz8 = {0, 0, 0, 0, 0, 0, 0, 0};
  __builtin_amdgcn_tensor_load_to_lds(g0, g1, gz4, gz4, gz8, 0);
}
#endif

__device__ __forceinline__ float bce_term(float p, int t, int c,
                                          float w_for_pos, float w_for_neg) {
  // BCE element: -log(p) if this channel is the label, else -log(1-p),
  // with torch-style clamp of the log argument to [1e-12, 1].
  float x = (t == c) ? p : (1.0f - p);
  x = fminf(fmaxf(x, 1e-12f), 1.0f);
  float l = -__logf(x);
  float w = (t == 0) ? w_for_neg
                     : ((t > 0 && t != 255) ? w_for_pos : 0.0f);
  return w * l;
}

// ---------------------------------------------------------------------------
// Kernel 0: zero the workspace (d_ws is poisoned, not re-zeroed by harness).
// ---------------------------------------------------------------------------
__global__ void ce_init(WS* ws) {
  ws->sum = 0.0;
  ws->pos = 0ull;
  ws->neg = 0ull;
  ws->maskc = 0ull;
}

// ---------------------------------------------------------------------------
// Kernel 1: global label counts (16 MB stream, int4 loads, block tree +
// one 64-bit atomic per counter per block).
// ---------------------------------------------------------------------------
__global__ __launch_bounds__(THREADS) void ce_count(const int* __restrict__ t,
                                                    WS* __restrict__ ws) {
  unsigned cp = 0, cn = 0, cm = 0;
  const int4* t4 = (const int4*)t;
  const int n4 = CE_NPIX / 4;
  for (int i = blockIdx.x * blockDim.x + threadIdx.x; i < n4;
       i += gridDim.x * blockDim.x) {
    int4 v = t4[i];
    cp += (v.x > 0); cn += (v.x == 0); cm += (v.x >= 0 && v.x != 255);
    cp += (v.y > 0); cn += (v.y == 0); cm += (v.y >= 0 && v.y != 255);
    cp += (v.z > 0); cn += (v.z == 0); cm += (v.z >= 0 && v.z != 255);
    cp += (v.w > 0); cn += (v.w == 0); cm += (v.w >= 0 && v.w != 255);
  }
  __shared__ unsigned sp[THREADS], sn[THREADS], sm[THREADS];
  const int tid = threadIdx.x;
  sp[tid] = cp; sn[tid] = cn; sm[tid] = cm;
  __syncthreads();
  for (int s = THREADS / 2; s > 0; s >>= 1) {
    if (tid < s) { sp[tid] += sp[tid + s]; sn[tid] += sn[tid + s]; sm[tid] += sm[tid + s]; }
    __syncthreads();
  }
  if (tid == 0) {
    atomicAdd(&ws->pos,   (unsigned long long)sp[0]);
    atomicAdd(&ws->neg,   (unsigned long long)sn[0]);
    atomicAdd(&ws->maskc, (unsigned long long)sm[0]);
  }
}

// ---------------------------------------------------------------------------
// Kernel 2: main weighted-BCE reduction. Each block owns K_TILES contiguous
// 4096-float tiles of predict. Tiles never cross a channel plane (4096 | 2^19),
// so (n, c) is per-tile scalar and matching target words are contiguous
// (L2-resident: 16 MB reused 19x). TDM double-buffers tiles into LDS; wave 0
// issues, everyone gates on s_wait_tensorcnt 0 + barrier.
// ---------------------------------------------------------------------------
__global__ __launch_bounds__(THREADS) void ce_main(const float* __restrict__ predict,
                                                   const int* __restrict__ target,
                                                   WS* __restrict__ ws) {
  __shared__ alignas(16) float smem[2 * TILE];   // 32 KB double buffer
  __shared__ float red[THREADS];
  const int tid = threadIdx.x;

  // class-balance weights from pass-1 counts (no gradient path; scalar)
  const double posn = (double)ws->pos;
  const double negn = (double)ws->neg;
  const double snum = posn + negn;
  const float w_for_pos = (float)(negn / snum);  // applied where t > 0
  const float w_for_neg = (float)(posn / snum);  // applied where t == 0

  float acc = 0.0f;
  const int tile0 = blockIdx.x * K_TILES;
  int buf = 0;

#if HAVE_TDM
  if ((tid >> 5) == 0) {  // wave 0 only (wave-uniform branch; TDM ignores EXEC)
    tdm_issue_1d((unsigned)(uintptr_t)&smem[0],
                 predict + (size_t)tile0 * TILE, TILE);
  }
#endif

  for (int k = 0; k < K_TILES; ++k) {
    const int base = (tile0 + k) * TILE;

#if HAVE_TDM
    __builtin_amdgcn_s_wait_tensorcnt(0);  // issuing wave blocks; others pass
    __syncthreads();                       // tile resident + prev buf released
    if ((k + 1) < K_TILES && (tid >> 5) == 0) {
      tdm_issue_1d((unsigned)(uintptr_t)&smem[(buf ^ 1) * TILE],
                   predict + (size_t)(base + TILE), TILE);  // prefetch next
    }
#else
    {   // fallback staging path: cooperative float4 copy into LDS
      const float4* gp = (const float4*)(predict + base);
      float4* dp = (float4*)&smem[buf * TILE];
      for (int i = tid; i < TILE / 4; i += THREADS) dp[i] = gp[i];
      __syncthreads();
    }
#endif

    // decode per-tile scalars: flat = ((n*C + c)*HW + r)
    const int q = base >> 19;            // base / HW
    const int n = q / CE_C;
    const int c = q - n * CE_C;
    const int pixbase = (n << 19) + (base & (CE_HW - 1));

    const float4* sp = (const float4*)&smem[buf * TILE];
    const int4*   tp = (const int4*)(target + pixbase);
    for (int i = tid; i < TILE / 4; i += THREADS) {
      float4 p = sp[i];      // ds_load_b128
      int4   t = tp[i];      // coalesced, L2-resident
      acc += bce_term(p.x, t.x, c, w_for_pos, w_for_neg);
      acc += bce_term(p.y, t.y, c, w_for_pos, w_for_neg);
      acc += bce_term(p.z, t.z, c, w_for_pos, w_for_neg);
      acc += bce_term(p.w, t.w, c, w_for_pos, w_for_neg);
    }
#if !HAVE_TDM
    __syncthreads();  // release buffer before next copy overwrites it
#endif
    buf ^= 1;
  }

  // block reduction: f32 tree in LDS, one f64 global atomic per block
  red[tid] = acc;
  __syncthreads();
  for (int s = THREADS / 2; s > 0; s >>= 1) {
    if (tid < s) red[tid] += red[tid + s];
    __syncthreads();
  }
  if (tid == 0) atomicAdd(&ws->sum, (double)red[0]);
}

// ---------------------------------------------------------------------------
// Kernel 3: finalize scalar: mean over kept elements = sum / (maskc * C).
// ---------------------------------------------------------------------------
__global__ void ce_final(const WS* __restrict__ ws, float* __restrict__ out) {
  out[0] = (float)(ws->sum / ((double)ws->maskc * (double)CE_C));
}

extern "C" void kernel_launch(void* const* d_in, const int* in_sizes, int n_in,
                              void* d_out, int out_size, void* d_ws, size_t ws_size,
                              hipStream_t stream) {
  const float* predict = (const float*)d_in[0];
  const int*   target  = (const int*)d_in[1];
  WS* ws = (WS*)d_ws;
  float* out = (float*)d_out;

  hipLaunchKernelGGL(ce_init,  dim3(1),           dim3(1),       0, stream, ws);
  hipLaunchKernelGGL(ce_count, dim3(1024),        dim3(THREADS), 0, stream, target, ws);
  hipLaunchKernelGGL(ce_main,  dim3(MAIN_BLOCKS), dim3(THREADS), 0, stream,
                     predict, target, ws);
  hipLaunchKernelGGL(ce_final, dim3(1),           dim3(1),       0, stream, ws, out);
}